// HierarchicalAttentionNetwork_90494960926749
// MI455X (gfx1250) — compile-verified
//
#include <hip/hip_runtime.h>
#include <hip/hip_bf16.h>
#include <math.h>

#define SEQ   512
#define BATCH 256
#define EMBD  200
#define HID   50
#define G3    150   // 3*HID
#define MLPD  100

typedef __attribute__((ext_vector_type(2))) float v2f;
typedef __attribute__((ext_vector_type(8))) float v8f;

__device__ __forceinline__ v8f wmma_f32(v2f a, v2f b, v8f c) {
    // (neg_a, A, neg_b, B, c_mod, C, reuse_a, reuse_b)
    return __builtin_amdgcn_wmma_f32_16x16x4_f32(false, a, false, b, (short)0, c, false, false);
}

__device__ __forceinline__ float sigmoidf_(float x) {
    return 1.0f / (1.0f + __expf(-x));
}

// Async global->LDS copies (gfx1250 ASYNCcnt path)
__device__ __forceinline__ void async_copy_b128(void* lds_ptr, const void* gptr) {
    unsigned lp = (unsigned)(unsigned long long)lds_ptr;
    asm volatile("global_load_async_to_lds_b128 %0, %1, off"
                 :: "v"(lp), "v"(gptr) : "memory");
}
__device__ __forceinline__ void async_copy_b64(void* lds_ptr, const void* gptr) {
    unsigned lp = (unsigned)(unsigned long long)lds_ptr;
    asm volatile("global_load_async_to_lds_b64 %0, %1, off"
                 :: "v"(lp), "v"(gptr) : "memory");
}
__device__ __forceinline__ void wait_async0() {
    asm volatile("s_wait_asynccnt 0" ::: "memory");
}

// ---------------------------------------------------------------------------
// Kernel 1: embedding gather + input projections gx = e @ Wih^T + bih (both dirs)
// 128 threads (4 waves), 64 rows (of S*B) per workgroup, 1 M-tile per wave.
// Embedding rows staged to LDS via async b128 copies.
// ---------------------------------------------------------------------------
__global__ __launch_bounds__(128) void k_proj(
    const int* __restrict__ tokens, const float* __restrict__ emb,
    const float* __restrict__ Wih_f, const float* __restrict__ bih_f,
    const float* __restrict__ Wih_b, const float* __restrict__ bih_b,
    float* __restrict__ gx_f, float* __restrict__ gx_b)
{
    __shared__ int   tok[64];
    __shared__ float e_lds[64 * EMBD];   // 51.2 KB

    const int tid = threadIdx.x;
    const int r0  = blockIdx.x * 64;

    if (tid < 64) tok[tid] = tokens[r0 + tid];
    __syncthreads();

    // 64 rows x 50 b128-chunks = 3200 chunks, 25 per thread, exact.
    for (int i = tid; i < 64 * (EMBD / 4); i += 128) {
        const int rl = i / (EMBD / 4), c4 = i - rl * (EMBD / 4);
        async_copy_b128(&e_lds[rl * EMBD + c4 * 4],
                        emb + (size_t)tok[rl] * EMBD + c4 * 4);
    }
    wait_async0();
    __syncthreads();

    const int w = tid >> 5, lane = tid & 31;
    const int mr = lane & 15, kh = lane >> 4;   // A: M=mr, K-half=kh
    const int arow = (w * 16 + mr) * EMBD;

    #pragma unroll 1
    for (int dir = 0; dir < 2; ++dir) {
        const float* Wih = dir ? Wih_b : Wih_f;
        const float* bih = dir ? bih_b : bih_f;
        float*       gx  = dir ? gx_b  : gx_f;

        // Clamped weight row pointers: out-of-range N columns compute garbage
        // into accumulator columns that are never stored.
        const float* wp[10];
        #pragma unroll
        for (int nt = 0; nt < 10; ++nt) {
            int n = nt * 16 + mr;
            if (n > G3 - 1) n = G3 - 1;
            wp[nt] = Wih + n * EMBD;
        }

        v8f zero = {};
        v8f acc[10];
        #pragma unroll
        for (int nt = 0; nt < 10; ++nt) acc[nt] = zero;

        for (int k0 = 0; k0 < EMBD; k0 += 4) {
            const int k = k0 + 2 * kh;
            v2f a;
            a.x = e_lds[arow + k];
            a.y = e_lds[arow + k + 1];
            #pragma unroll
            for (int nt = 0; nt < 10; ++nt) {
                v2f bf;
                bf.x = wp[nt][k];
                bf.y = wp[nt][k + 1];
                acc[nt] = wmma_f32(a, bf, acc[nt]);
            }
        }
        #pragma unroll
        for (int nt = 0; nt < 10; ++nt) {
            const int n = nt * 16 + mr;
            if (n < G3) {
                const float bv = bih[n];
                #pragma unroll
                for (int r = 0; r < 8; ++r) {
                    const int row = r0 + w * 16 + r + 8 * kh;
                    gx[(size_t)row * G3 + n] = acc[nt][r] + bv;
                }
            }
        }
    }
}

// ---------------------------------------------------------------------------
// Kernel 2: GRU scan. 8 blocks = {fwd,bwd} x 4 batch-slices of 64.
// 256 threads (8 waves): wave w -> M-tile (w>>1), N-tile group (w&1)*5..+4.
// h-state + gh buffer in LDS; Whh^T fragments pre-packed in VGPRs; per-step
// gx loads issued BEFORE the WMMA phase so memory latency hides under matrix
// work; bhh biases preloaded to registers.
// ---------------------------------------------------------------------------
__global__ __launch_bounds__(256) void k_gru(
    const float* __restrict__ gx_f, const float* __restrict__ gx_b,
    const float* __restrict__ Whh_f, const float* __restrict__ bhh_f,
    const float* __restrict__ Whh_b, const float* __restrict__ bhh_b,
    float* __restrict__ h_f, float* __restrict__ h_b)
{
    __shared__ float h_lds[64 * 52];     // 13.3 KB, cols 50..51 stay zero (K pad)
    __shared__ float gh_lds[64 * 160];   // 41 KB

    const int tid = threadIdx.x;
    const int bx  = blockIdx.x;
    const int dir = bx >> 2;
    const int b0  = (bx & 3) * 64;

    const float* gx   = dir ? gx_b  : gx_f;
    const float* Whh  = dir ? Whh_b : Whh_f;
    const float* bhh  = dir ? bhh_b : bhh_f;
    float*       hout = dir ? h_b   : h_f;

    for (int i = tid; i < 64 * 52; i += 256) h_lds[i] = 0.0f;

    const int w = tid >> 5, lane = tid & 31;
    const int mr = lane & 15, kh = lane >> 4;
    const int mt = w >> 1;        // M-tile 0..3
    const int ng = w & 1;         // N-tile group
    const int arow = (mt * 16 + mr) * 52;

    // Pre-pack B = Whh^T fragments: constant across all 512 steps.
    // Clamped indices: K>=50 pairs against zero-padded A columns (0 * x = 0);
    // N>=150 columns land in unread gh_lds columns.
    v2f bfrag[5][13];
    #pragma unroll
    for (int i2 = 0; i2 < 5; ++i2) {
        int n = (ng * 5 + i2) * 16 + mr;
        if (n > G3 - 1) n = G3 - 1;
        #pragma unroll
        for (int kk = 0; kk < 13; ++kk) {
            int k = kk * 4 + 2 * kh;
            if (k > HID - 2) k = HID - 2;
            v2f bf;
            bf.x = Whh[n * HID + k];
            bf.y = Whh[n * HID + k + 1];
            bfrag[i2][kk] = bf;
        }
    }

    // Preload loop-invariant gate biases for this thread's elementwise items.
    float br[13], bz[13], bn[13];
    #pragma unroll
    for (int it = 0; it < 13; ++it) {
        int i = tid + it * 256;
        if (i > 64 * HID - 1) i = 64 * HID - 1;
        const int j = i % HID;
        br[it] = bhh[j];
        bz[it] = bhh[HID + j];
        bn[it] = bhh[2 * HID + j];
    }
    __syncthreads();

    const v8f zero = {};
    for (int step = 0; step < SEQ; ++step) {
        const int t = dir ? (SEQ - 1 - step) : step;
        const size_t base  = ((size_t)t * BATCH + b0) * G3;
        const size_t hbase = ((size_t)t * BATCH + b0) * HID;

        // Issue this step's gx loads early; they overlap the WMMA phase.
        float xr[13], xz[13], xn[13];
        #pragma unroll
        for (int it = 0; it < 13; ++it) {
            int i = tid + it * 256;
            if (i > 64 * HID - 1) i = 64 * HID - 1;   // clamped duplicates, unused
            const int bl = i / HID, j = i - bl * HID;
            const size_t idx = base + (size_t)bl * G3 + j;
            xr[it] = gx[idx];
            xz[it] = gx[idx + HID];
            xn[it] = gx[idx + 2 * HID];
        }

        // gh = h @ Whh^T  (16x16x4 f32 WMMA, A from LDS, B from registers)
        v8f acc[5];
        #pragma unroll
        for (int i2 = 0; i2 < 5; ++i2) acc[i2] = zero;
        #pragma unroll
        for (int kk = 0; kk < 13; ++kk) {
            const int k = kk * 4 + 2 * kh;
            v2f a;
            a.x = h_lds[arow + k];
            a.y = h_lds[arow + k + 1];
            #pragma unroll
            for (int i2 = 0; i2 < 5; ++i2)
                acc[i2] = wmma_f32(a, bfrag[i2][kk], acc[i2]);
        }
        #pragma unroll
        for (int i2 = 0; i2 < 5; ++i2) {
            const int n = (ng * 5 + i2) * 16 + mr;
            #pragma unroll
            for (int r = 0; r < 8; ++r)
                gh_lds[(mt * 16 + r + 8 * kh) * 160 + n] = acc[i2][r];
        }
        __syncthreads();

        // gate update (registers + LDS only on the critical path)
        #pragma unroll
        for (int it = 0; it < 13; ++it) {
            const int i = tid + it * 256;
            if (i > 64 * HID - 1) continue;
            const int bl = i / HID, j = i - bl * HID;
            const float ghr = gh_lds[bl * 160 + j]           + br[it];
            const float ghz = gh_lds[bl * 160 + HID + j]     + bz[it];
            const float ghn = gh_lds[bl * 160 + 2 * HID + j] + bn[it];
            const float rg = sigmoidf_(xr[it] + ghr);
            const float zg = sigmoidf_(xz[it] + ghz);
            const float ngate = tanhf(xn[it] + rg * ghn);
            const float hnew = (1.0f - zg) * ngate + zg * h_lds[bl * 52 + j];
            h_lds[bl * 52 + j] = hnew;
            hout[hbase + (size_t)bl * HID + j] = hnew;
        }
        // prefetch next step's gx slice into cache (global_prefetch_b8)
        if (step + 1 < SEQ) {
            const int tn = dir ? (SEQ - 2 - step) : (step + 1);
            __builtin_prefetch(gx + ((size_t)tn * BATCH + b0) * G3 + tid * 37, 0, 0);
        }
        __syncthreads();
    }
}

// ---------------------------------------------------------------------------
// Kernel 3: scores[s,b] = tanh(h @ W_mlp^T + b_mlp) @ ctx
// 256 threads (8 waves), 128 rows per WG, 1 M-tile per wave, K=100, N=100.
// h tiles staged to LDS via async b64 copies.
// ---------------------------------------------------------------------------
__global__ __launch_bounds__(256) void k_score(
    const float* __restrict__ h_f, const float* __restrict__ h_b,
    const float* __restrict__ W_mlp, const float* __restrict__ b_mlp,
    const float* __restrict__ ctx, float* __restrict__ scores)
{
    __shared__ float hbuf[128 * 104];    // 53.2 KB
    __shared__ float sred[128];

    const int tid = threadIdx.x;
    const int r0  = blockIdx.x * 128;

    // 128 rows x 50 b64-chunks = 6400, 25 per thread, exact.
    for (int i = tid; i < 128 * 50; i += 256) {
        const int rl = i / 50, p = i - rl * 50;
        const float* src = (p < 25)
            ? (h_f + (size_t)(r0 + rl) * HID + 2 * p)
            : (h_b + (size_t)(r0 + rl) * HID + 2 * (p - 25));
        async_copy_b64(&hbuf[rl * 104 + 2 * p], src);
    }
    if (tid < 128) sred[tid] = 0.0f;
    wait_async0();
    __syncthreads();

    const int w = tid >> 5, lane = tid & 31;
    const int mr = lane & 15, kh = lane >> 4;
    const int arow = (w * 16 + mr) * 104;

    const float* wp[7];
    #pragma unroll
    for (int nt = 0; nt < 7; ++nt) {
        int n = nt * 16 + mr;
        if (n > MLPD - 1) n = MLPD - 1;
        wp[nt] = W_mlp + n * MLPD;
    }

    const v8f zero = {};
    v8f acc[7];
    #pragma unroll
    for (int nt = 0; nt < 7; ++nt) acc[nt] = zero;

    for (int kk = 0; kk < 25; ++kk) {
        const int k = kk * 4 + 2 * kh;
        v2f a;
        a.x = hbuf[arow + k];
        a.y = hbuf[arow + k + 1];
        #pragma unroll
        for (int nt = 0; nt < 7; ++nt) {
            v2f bf;
            bf.x = wp[nt][k];
            bf.y = wp[nt][k + 1];
            acc[nt] = wmma_f32(a, bf, acc[nt]);
        }
    }

    float part[8] = {0, 0, 0, 0, 0, 0, 0, 0};
    #pragma unroll
    for (int nt = 0; nt < 7; ++nt) {
        const int n = nt * 16 + mr;
        if (n < MLPD) {
            const float cx = ctx[n], bm = b_mlp[n];
            #pragma unroll
            for (int r = 0; r < 8; ++r)
                part[r] += tanhf(acc[nt][r] + bm) * cx;
        }
    }
    #pragma unroll
    for (int r = 0; r < 8; ++r)
        atomicAdd(&sred[w * 16 + r + 8 * kh], part[r]);
    __syncthreads();

    if (tid < 128) scores[r0 + tid] = sred[tid];
}

// ---------------------------------------------------------------------------
// Kernel 4: softmax over S per batch element + weighted sum -> doc (B, 2H)
// ---------------------------------------------------------------------------
__global__ __launch_bounds__(256) void k_doc(
    const float* __restrict__ scores,
    const float* __restrict__ h_f, const float* __restrict__ h_b,
    float* __restrict__ out)
{
    __shared__ float alpha[SEQ];
    __shared__ float red[256];

    const int b = blockIdx.x;
    const int tid = threadIdx.x;

    float m = -1e30f;
    for (int s = tid; s < SEQ; s += 256) m = fmaxf(m, scores[s * BATCH + b]);
    red[tid] = m;
    __syncthreads();
    for (int off = 128; off > 0; off >>= 1) {
        if (tid < off) red[tid] = fmaxf(red[tid], red[tid + off]);
        __syncthreads();
    }
    const float maxv = red[0];
    __syncthreads();

    float p = 0.0f;
    for (int s = tid; s < SEQ; s += 256) {
        const float e = __expf(scores[s * BATCH + b] - maxv);
        alpha[s] = e;
        p += e;
    }
    red[tid] = p;
    __syncthreads();
    for (int off = 128; off > 0; off >>= 1) {
        if (tid < off) red[tid] += red[tid + off];
        __syncthreads();
    }
    const float inv = 1.0f / red[0];

    if (tid < 2 * HID) {
        float accd = 0.0f;
        for (int s = 0; s < SEQ; ++s) {
            const float hv = (tid < HID)
                ? h_f[((size_t)s * BATCH + b) * HID + tid]
                : h_b[((size_t)s * BATCH + b) * HID + (tid - HID)];
            accd += alpha[s] * hv;
        }
        out[b * (2 * HID) + tid] = accd * inv;
    }
}

// ---------------------------------------------------------------------------
extern "C" void kernel_launch(void* const* d_in, const int* in_sizes, int n_in,
                              void* d_out, int out_size, void* d_ws, size_t ws_size,
                              hipStream_t stream) {
    const int*   tokens = (const int*)  d_in[0];
    const float* emb    = (const float*)d_in[1];
    const float* Wih_f  = (const float*)d_in[2];
    const float* Whh_f  = (const float*)d_in[3];
    const float* bih_f  = (const float*)d_in[4];
    const float* bhh_f  = (const float*)d_in[5];
    const float* Wih_b  = (const float*)d_in[6];
    const float* Whh_b  = (const float*)d_in[7];
    const float* bih_b  = (const float*)d_in[8];
    const float* bhh_b  = (const float*)d_in[9];
    const float* W_mlp  = (const float*)d_in[10];
    const float* b_mlp  = (const float*)d_in[11];
    const float* ctx    = (const float*)d_in[12];
    float* out = (float*)d_out;

    float* ws = (float*)d_ws;
    const size_t ngx = (size_t)SEQ * BATCH * G3;
    const size_t nh  = (size_t)SEQ * BATCH * HID;
    float* gx_f   = ws;
    float* gx_b   = gx_f + ngx;
    float* h_f    = gx_b + ngx;
    float* h_b    = h_f + nh;
    float* scores = h_b + nh;

    k_proj<<<dim3((SEQ * BATCH) / 64), dim3(128), 0, stream>>>(
        tokens, emb, Wih_f, bih_f, Wih_b, bih_b, gx_f, gx_b);

    k_gru<<<dim3(8), dim3(256), 0, stream>>>(
        gx_f, gx_b, Whh_f, bhh_f, Whh_b, bhh_b, h_f, h_b);

    k_score<<<dim3((SEQ * BATCH) / 128), dim3(256), 0, stream>>>(
        h_f, h_b, W_mlp, b_mlp, ctx, scores);

    k_doc<<<dim3(BATCH), dim3(256), 0, stream>>>(scores, h_f, h_b, out);
}